// LightGCN_80350248174012
// MI455X (gfx1250) — compile-verified
//
#include <hip/hip_runtime.h>
#include <stdint.h>

#define N_USERS 100000
#define N_ITEMS 200000
#define N_NODES (N_USERS + N_ITEMS)
#define EMB     64
#define NNZ     9600000
#define BATCH   4096
#define LAYERS  3
#define TILE    1024              // edges staged per block (NNZ = 9375 * 1024 exactly)
#define BLK     256               // 8 waves of 32
#define SCAN_CHUNK 2048
#define NSCAN   ((N_NODES + SCAN_CHUNK - 1) / SCAN_CHUNK)   // 147

// ---------------- CDNA5 async global->LDS staging (probe-confirmed builtin) ----------------
typedef int v4i __attribute__((vector_size(16)));
typedef __attribute__((address_space(1))) v4i g_v4i;
typedef __attribute__((address_space(3))) v4i l_v4i;

__device__ __forceinline__ void async_copy16(void* lds_dst, const void* gsrc) {
#if defined(__HIP_DEVICE_COMPILE__) && __has_builtin(__builtin_amdgcn_global_load_async_to_lds_b128)
    __builtin_amdgcn_global_load_async_to_lds_b128(
        (g_v4i*)(uintptr_t)gsrc, (l_v4i*)(uint32_t)(uintptr_t)lds_dst, 0, 0);
#else
    asm volatile("global_load_async_to_lds_b128 %0, %1, off"
                 :: "v"((uint32_t)(uintptr_t)lds_dst), "v"(gsrc) : "memory");
#endif
}

__device__ __forceinline__ void wait_async0() {
#if defined(__HIP_DEVICE_COMPILE__) && __has_builtin(__builtin_amdgcn_s_wait_asynccnt)
    __builtin_amdgcn_s_wait_asynccnt(0);
#else
    asm volatile("s_wait_asynccnt 0" ::: "memory");
#endif
}

// ---------------- CSR SpMM: one wave per row, zero atomics (THE hot kernel) ----------------
// Row index is wave-uniform (readfirstlane), so row_ptr and the packed
// (col,val) edge records scalarize to s_load_b64 through the constant cache;
// lanes cover the 64-float row as float2 (256B coalesced gather, L2-resident);
// one non-atomic global_store_b64 per lane at the end. No zero-fill needed.
__global__ void __launch_bounds__(BLK) k_spmm_csr(const float* __restrict__ h_in,
                                                  float*       __restrict__ h_out,
                                                  const uint32_t* __restrict__ row_ptr,
                                                  const uint2* __restrict__ csr_ev) {
    const int lane = threadIdx.x & 31;
    const int wv   = __builtin_amdgcn_readfirstlane(threadIdx.x >> 5);   // uniform
    const int row  = blockIdx.x * (BLK / 32) + wv;                       // uniform, < N_NODES
    const int jBeg = (int)row_ptr[row];                                  // s_load
    const int jEnd = (int)row_ptr[row + 1];
    float2 acc = make_float2(0.f, 0.f);
#pragma unroll 4
    for (int j = jBeg; j < jEnd; ++j) {
        const uint2 ev = csr_ev[j];                                      // s_load_b64 (uniform)
        const int   c  = (int)ev.x;
        const float v  = __uint_as_float(ev.y);
        const float2 x = *(const float2*)(h_in + ((size_t)c << 6) + (lane << 1));
        acc.x += v * x.x;
        acc.y += v * x.y;
    }
    *(float2*)(h_out + ((size_t)row << 6) + (lane << 1)) = acc;
}

// scatter COO -> packed CSR; COO tile staged into LDS via CDNA5 async global->LDS b128
__global__ void __launch_bounds__(BLK) k_scatter(const int* __restrict__ rows,
                                                 const int* __restrict__ cols,
                                                 const float* __restrict__ vals,
                                                 uint32_t* __restrict__ cursor,
                                                 uint2* __restrict__ csr_ev) {
    __shared__ __align__(16) int   s_rows[TILE];
    __shared__ __align__(16) int   s_cols[TILE];
    __shared__ __align__(16) float s_vals[TILE];
    const long base = (long)blockIdx.x * TILE;
    const int  t = threadIdx.x;
    async_copy16(&s_rows[t * 4], rows + base + t * 4);
    async_copy16(&s_cols[t * 4], cols + base + t * 4);
    async_copy16(&s_vals[t * 4], vals + base + t * 4);
    wait_async0();
    __syncthreads();
    for (int k = t * 4; k < t * 4 + 4; ++k) {
        int r = s_rows[k];
        uint32_t p = atomicAdd(&cursor[r], 1u);
        csr_ev[p] = make_uint2((uint32_t)s_cols[k], __float_as_uint(s_vals[k]));
    }
}

// ---------------- feature kernels ----------------

// node_feat = concat(user_table, item_table), float4-vectorized
__global__ void k_concat(const float4* __restrict__ ut, const float4* __restrict__ it,
                         float4* __restrict__ dst) {
    int i = blockIdx.x * blockDim.x + threadIdx.x;        // i < N_NODES*16
    dst[i] = (i < N_USERS * 16) ? ut[i] : it[i - N_USERS * 16];
}

// out[i] (+)= 0.25 * h[node]  for the three gathered index streams
__global__ void k_gather(const float* __restrict__ h,
                         const int* __restrict__ user,
                         const int* __restrict__ pos,
                         const int* __restrict__ neg,
                         float2* __restrict__ out, int accumulate) {
    int i = blockIdx.x * blockDim.x + threadIdx.x;        // i < 3*BATCH*32
    int d     = i & 31;
    int b     = (i >> 5) & (BATCH - 1);
    int which = i >> 17;                                  // 5 + log2(4096)
    int node = (which == 0) ? user[b] : N_USERS + ((which == 1) ? pos[b] : neg[b]);
    float2 x = *(const float2*)(h + ((size_t)node << 6) + (d << 1));
    x.x *= 0.25f; x.y *= 0.25f;
    if (accumulate) { float2 o = out[i]; x.x += o.x; x.y += o.y; }
    out[i] = x;
}

// ---------------- CSR construction (per launch; graph-replay safe) ----------------

__global__ void k_zero_u32(uint32_t* __restrict__ p, int n) {
    int i = blockIdx.x * blockDim.x + threadIdx.x;
    if (i < n) p[i] = 0u;
}

// per-row edge counts (integer atomics, L2-resident 1.2MB table)
__global__ void k_hist(const int4* __restrict__ rows4, uint32_t* __restrict__ counts) {
    int i = blockIdx.x * blockDim.x + threadIdx.x;        // i < NNZ/4
    int4 r = rows4[i];
    atomicAdd(&counts[r.x], 1u);
    atomicAdd(&counts[r.y], 1u);
    atomicAdd(&counts[r.z], 1u);
    atomicAdd(&counts[r.w], 1u);
}

__global__ void k_chunk_sum(const uint32_t* __restrict__ counts, uint32_t* __restrict__ bsum) {
    __shared__ uint32_t s[BLK];
    int b = blockIdx.x, t = threadIdx.x;
    long base = (long)b * SCAN_CHUNK;
    uint32_t sum = 0;
    for (int k = t; k < SCAN_CHUNK; k += BLK) {
        long i = base + k;
        sum += (i < N_NODES) ? counts[i] : 0u;
    }
    s[t] = sum; __syncthreads();
    for (int off = BLK / 2; off > 0; off >>= 1) {
        if (t < off) s[t] += s[t + off];
        __syncthreads();
    }
    if (t == 0) bsum[b] = s[0];
}

__global__ void k_scan_partials(uint32_t* __restrict__ bsum, int n) {
    if (blockIdx.x == 0 && threadIdx.x == 0) {
        uint32_t run = 0;
        for (int i = 0; i < n; ++i) { uint32_t v = bsum[i]; bsum[i] = run; run += v; }
    }
}

__global__ void k_chunk_scan(const uint32_t* __restrict__ counts,
                             const uint32_t* __restrict__ bsum,
                             uint32_t* __restrict__ row_ptr,
                             uint32_t* __restrict__ cursor) {
    __shared__ uint32_t s[BLK];
    const int PER = SCAN_CHUNK / BLK;                     // 8
    int b = blockIdx.x, t = threadIdx.x;
    long base = (long)b * SCAN_CHUNK;
    uint32_t loc[PER], tsum = 0;
    for (int k = 0; k < PER; ++k) {
        long i = base + t * PER + k;
        uint32_t v = (i < N_NODES) ? counts[i] : 0u;
        loc[k] = tsum;                                    // thread-local exclusive
        tsum += v;
    }
    s[t] = tsum; __syncthreads();
    for (int off = 1; off < BLK; off <<= 1) {             // Hillis-Steele inclusive
        uint32_t v = s[t] + ((t >= off) ? s[t - off] : 0u);
        __syncthreads();
        s[t] = v;
        __syncthreads();
    }
    uint32_t texcl = s[t] - tsum + bsum[b];
    for (int k = 0; k < PER; ++k) {
        long i = base + t * PER + k;
        if (i < N_NODES) { uint32_t p = texcl + loc[k]; row_ptr[i] = p; cursor[i] = p; }
    }
    if (b == 0 && t == 0) row_ptr[N_NODES] = NNZ;
}

// ---------------- launcher ----------------
extern "C" void kernel_launch(void* const* d_in, const int* in_sizes, int n_in,
                              void* d_out, int out_size, void* d_ws, size_t ws_size,
                              hipStream_t stream) {
    const int*   user = (const int*)d_in[0];
    const int*   pos  = (const int*)d_in[1];
    const int*   neg  = (const int*)d_in[2];
    const float* ut   = (const float*)d_in[3];
    const float* it   = (const float*)d_in[4];
    const int*   rows = (const int*)d_in[5];
    const int*   cols = (const int*)d_in[6];
    const float* vals = (const float*)d_in[7];

    // workspace layout (all chunks 16B-aligned in order)
    char* w = (char*)d_ws;
    float*    bufA    = (float*)w;    w += (size_t)N_NODES * EMB * 4;   // 76.8 MB
    float*    bufB    = (float*)w;    w += (size_t)N_NODES * EMB * 4;   // 76.8 MB
    uint2*    csr_ev  = (uint2*)w;    w += (size_t)NNZ * 8;             // 76.8 MB packed (col,val)
    uint32_t* counts  = (uint32_t*)w; w += (size_t)N_NODES * 4;         // 1.2 MB
    uint32_t* row_ptr = (uint32_t*)w; w += (size_t)(N_NODES + 4) * 4;   // 1.2 MB
    uint32_t* cursor  = (uint32_t*)w; w += (size_t)N_NODES * 4;         // 1.2 MB
    uint32_t* bsum    = (uint32_t*)w;                                   // 147 u32

    float2* out = (float2*)d_out;

    const int nFeat4  = N_NODES * (EMB / 4);        // 4,800,000 float4
    const int nGather = 3 * BATCH * (EMB / 2);      // 393,216 float2
    const int nTiles  = NNZ / TILE;                 // 9,375

    // layer 0: node features + their gather contribution
    k_concat<<<nFeat4 / BLK, BLK, 0, stream>>>((const float4*)ut, (const float4*)it,
                                               (float4*)bufA);
    k_gather<<<nGather / BLK, BLK, 0, stream>>>(bufA, user, pos, neg, out, 0);

    // build CSR (fresh every call: counts zeroed, cursor rewritten by scan)
    k_zero_u32<<<(N_NODES + BLK - 1) / BLK, BLK, 0, stream>>>(counts, N_NODES);
    k_hist<<<(NNZ / 4) / BLK, BLK, 0, stream>>>((const int4*)rows, counts);
    k_chunk_sum<<<NSCAN, BLK, 0, stream>>>(counts, bsum);
    k_scan_partials<<<1, BLK, 0, stream>>>(bsum, NSCAN);
    k_chunk_scan<<<NSCAN, BLK, 0, stream>>>(counts, bsum, row_ptr, cursor);
    k_scatter<<<nTiles, BLK, 0, stream>>>(rows, cols, vals, cursor, csr_ev);

    // 3 propagation layers: wave-per-row CSR SpMM (no atomics, no zeroing)
    float* hin  = bufA;
    float* hout = bufB;
    for (int l = 0; l < LAYERS; ++l) {
        k_spmm_csr<<<N_NODES / (BLK / 32), BLK, 0, stream>>>(hin, hout, row_ptr, csr_ev);
        k_gather<<<nGather / BLK, BLK, 0, stream>>>(hout, user, pos, neg, out, 1);
        float* tmp = hin; hin = hout; hout = tmp;
    }
}